// Hdsdm_57904749084788
// MI455X (gfx1250) — compile-verified
//
#include <hip/hip_runtime.h>
#include <hip/hip_bf16.h>

typedef float v2f __attribute__((ext_vector_type(2)));
typedef float v8f __attribute__((ext_vector_type(8)));

// Workspace layout (in floats)
#define WS_XP    0                   // 128*128  projected X
#define WS_Y2    (128*128)           // 128      ||Xp_b||^2
#define WS_NUM   (WS_Y2 + 128)       // 128*112  unnormalized numerator (padded NC)
#define WS_DEN   (WS_NUM + 128*112)  // 128      softmax denominator

#define LDS_STRIDE 132               // 128 + 4: float4-aligned, kills bank conflicts

// ---------------- Projection of X onto the ball + y2 ----------------
__global__ void proj_kernel(const float* __restrict__ X, float* __restrict__ ws) {
    __shared__ float red[128];
    __shared__ float scale_s;
    int b = blockIdx.x, d = threadIdx.x;
    float x = X[b * 128 + d];
    red[d] = x * x;
    __syncthreads();
    for (int s = 64; s > 0; s >>= 1) {
        if (d < s) red[d] += red[d + s];
        __syncthreads();
    }
    if (d == 0) {
        float norm = fmaxf(sqrtf(red[0]), 1e-15f);
        const float maxn = 1.0f - 4e-3f;
        scale_s = (norm > maxn) ? (maxn / norm) : 1.0f;
        float cn = fminf(norm, maxn);
        ws[WS_Y2 + b] = cn * cn;
    }
    __syncthreads();
    ws[WS_XP + b * 128 + d] = x * scale_s;
}

// ---------------- Fused: x2 + GEMM1 (A.Xp^T) -> weights -> GEMM2 (W^T.C) ----------------
__global__ __launch_bounds__(256) void hyp_main_kernel(const float* __restrict__ A,
                                                       const float* __restrict__ Cw,
                                                       float* __restrict__ ws) {
    __shared__ float ldsA[128 * LDS_STRIDE];   // A tile (phase 1), ~66 KB
    __shared__ float ldsX[128 * LDS_STRIDE];   // Xp (phase 1) then W (phase 2)
    __shared__ float x2s[128];                 // ||A_row||^2 for this tile

    const int tid    = threadIdx.x;
    const int lane   = tid & 31;
    const int wave   = tid >> 5;       // 0..7
    const int lane15 = lane & 15;
    const int hi     = lane >> 4;      // 0 (K=k,k+1) or 1 (K=k+2,k+3)
    const int nb     = blockIdx.x;     // prototype tile: rows nb*128 .. +127

    // ---- Stage A tile and Xp into LDS (coalesced float4) ----
    const float4* Ag = reinterpret_cast<const float4*>(A + (size_t)nb * 128 * 128);
    const float4* Xg = reinterpret_cast<const float4*>(ws + WS_XP);
    for (int idx = tid; idx < 128 * 32; idx += 256) {
        const int row = idx >> 5, c4 = idx & 31;
        *reinterpret_cast<float4*>(&ldsA[row * LDS_STRIDE + c4 * 4]) = Ag[idx];
        *reinterpret_cast<float4*>(&ldsX[row * LDS_STRIDE + c4 * 4]) = Xg[idx];
    }
    __syncthreads();

    // ---- x2 for this tile's rows (fused; avoids a second HBM pass over A) ----
    if (tid < 128) {
        float s = 0.f;
#pragma unroll 8
        for (int d = 0; d < 128; ++d) {
            const float v = ldsA[tid * LDS_STRIDE + d];
            s += v * v;
        }
        x2s[tid] = s;
    }
    __syncthreads();

    // ---- Phase 1: S = A_tile . Xp^T via v_wmma_f32_16x16x4_f32 ----
    const int m_loc = wave * 16;                   // this wave's 16 rows (local)
    v8f acc[8] = {};                               // 8 column tiles of b (128 cols)

    for (int k = 0; k < 128; k += 4) {
        const int kk = k + (hi << 1);
        v2f a;
        a.x = ldsA[(m_loc + lane15) * LDS_STRIDE + kk];
        a.y = ldsA[(m_loc + lane15) * LDS_STRIDE + kk + 1];
#pragma unroll
        for (int j = 0; j < 8; ++j) {
            const int bcol = j * 16 + lane15;
            v2f bf;
            bf.x = ldsX[bcol * LDS_STRIDE + kk];
            bf.y = ldsX[bcol * LDS_STRIDE + kk + 1];
            acc[j] = __builtin_amdgcn_wmma_f32_16x16x4_f32(
                false, a, false, bf, (short)0, acc[j], false, false);
        }
    }

    // ---- Elementwise: dot -> Poincare distance -> softmin weight ----
    float x2v[8];
#pragma unroll
    for (int r = 0; r < 8; ++r) x2v[r] = x2s[m_loc + r + hi * 8];

#pragma unroll
    for (int j = 0; j < 8; ++j) {
        const float y2v = ws[WS_Y2 + j * 16 + lane15];
#pragma unroll
        for (int r = 0; r < 8; ++r) {
            const float s     = acc[j][r];           // <A_n, Xp_b>
            const float x2    = x2v[r];
            const float alpha = 1.0f - 2.0f * s + y2v;       // 1 + 2xy + y2 (xy = -s)
            const float beta  = 1.0f - x2;
            float num2 = alpha * alpha * x2 - 2.0f * alpha * beta * s + beta * beta * y2v;
            float den  = fmaxf(1.0f - 2.0f * s + x2 * y2v, 1e-15f);
            float z    = sqrtf(fmaxf(num2, 0.0f)) / den;     // ||mobius_add(-A, Xp)||
            z = fminf(z, 1.0f - 1e-7f);
            // exp(-2*artanh(z)/4) == ((1-z)/(1+z))^(1/4)
            acc[j][r] = sqrtf(sqrtf((1.0f - z) / (1.0f + z)));
        }
    }

    __syncthreads();   // everyone done reading ldsX (Xp) / ldsA

    // ---- Restage weights: W[m_local][b] (reuse ldsX) ----
#pragma unroll
    for (int j = 0; j < 8; ++j) {
#pragma unroll
        for (int r = 0; r < 8; ++r) {
            const int m_local = m_loc + r + hi * 8;
            const int bcol    = j * 16 + lane15;
            ldsX[m_local * LDS_STRIDE + bcol] = acc[j][r];
        }
    }
    __syncthreads();

    // ---- Softmax denominator partial: one column per thread (first 128) ----
    if (tid < 128) {
        float s = 0.f;
        for (int m = 0; m < 128; ++m) s += ldsX[m * LDS_STRIDE + tid];
        atomicAdd(&ws[WS_DEN + tid], s);
    }

    // ---- Phase 2: out_part[b][c] = W^T . C_tile, K = 128 (m) ----
    const int b_base = wave * 16;                  // this wave's 16 b-rows
    v8f acc2[7] = {};                              // 7 tiles cover NC=100 (pad 112)

    const float* Ct = Cw + (size_t)nb * 128 * 100; // this block's 128 rows of C
    for (int k = 0; k < 128; k += 4) {
        const int kk = k + (hi << 1);
        v2f a;                                      // (W^T)[b][m] = W[m][b]
        a.x = ldsX[kk * LDS_STRIDE + b_base + lane15];
        a.y = ldsX[(kk + 1) * LDS_STRIDE + b_base + lane15];
#pragma unroll
        for (int j = 0; j < 7; ++j) {
            const int ccol = j * 16 + lane15;
            const bool ok  = (ccol < 100);
            v2f bf;
            bf.x = ok ? Ct[(size_t)kk * 100 + ccol]       : 0.0f;
            bf.y = ok ? Ct[(size_t)(kk + 1) * 100 + ccol] : 0.0f;
            acc2[j] = __builtin_amdgcn_wmma_f32_16x16x4_f32(
                false, a, false, bf, (short)0, acc2[j], false, false);
        }
    }

    // ---- Accumulate numerator partials (region is L2-resident; not HBM traffic) ----
#pragma unroll
    for (int j = 0; j < 7; ++j) {
#pragma unroll
        for (int r = 0; r < 8; ++r) {
            const int brow = b_base + r + hi * 8;
            const int ccol = j * 16 + lane15;
            atomicAdd(&ws[WS_NUM + brow * 112 + ccol], acc2[j][r]);
        }
    }
}

// ---------------- Normalize ----------------
__global__ void fin_kernel(const float* __restrict__ ws, float* __restrict__ out) {
    int i = blockIdx.x * blockDim.x + threadIdx.x;
    if (i >= 128 * 100) return;
    int b = i / 100, c = i % 100;
    out[i] = ws[WS_NUM + b * 112 + c] / ws[WS_DEN + b];
}

extern "C" void kernel_launch(void* const* d_in, const int* in_sizes, int n_in,
                              void* d_out, int out_size, void* d_ws, size_t ws_size,
                              hipStream_t stream) {
    const float* X  = (const float*)d_in[0];   // [128,128]
    const float* A  = (const float*)d_in[1];   // [16384,1,128]
    const float* Cw = (const float*)d_in[2];   // [16384,100]
    float* out = (float*)d_out;
    float* ws  = (float*)d_ws;

    // zero numerator + denominator accumulators (contiguous)
    hipMemsetAsync(ws + WS_NUM, 0, (128 * 112 + 128) * sizeof(float), stream);

    proj_kernel<<<128, 128, 0, stream>>>(X, ws);
    hyp_main_kernel<<<128, 256, 0, stream>>>(A, Cw, ws);
    fin_kernel<<<(128 * 100 + 255) / 256, 256, 0, stream>>>(ws, out);
}